// NodeConvolution_83786222011240
// MI455X (gfx1250) — compile-verified
//
#include <hip/hip_runtime.h>

typedef float v2f __attribute__((ext_vector_type(2)));
typedef float v8f __attribute__((ext_vector_type(8)));

#define N_NODES 50000
#define F_NODE  256
#define F_HEDGE 128
#define E1      800000
#define E2      400000
#define CHUNK   128

// ---------------------------------------------------------------------------
// Zero-fill workspace (agg buffers + csum vectors), float4-wide.
// ---------------------------------------------------------------------------
__global__ void zero_kernel(float4* __restrict__ p, int n4) {
    int i = blockIdx.x * blockDim.x + threadIdx.x;
    if (i < n4) p[i] = make_float4(0.f, 0.f, 0.f, 0.f);
}

// ---------------------------------------------------------------------------
// Segment-sum of scaled sender features, exploiting sorted receivers.
// blockDim.x == F. Each block handles CHUNK contiguous edges; each thread owns
// one feature column and accumulates in a register across runs of equal
// receiver, flushing with atomicAdd only at run boundaries.
// Thread 0 additionally accumulates sum of conv weights per receiver (csum).
// ---------------------------------------------------------------------------
template <int F>
__global__ void edge_agg_kernel(const float* __restrict__ feats,
                                const int*   __restrict__ senders,
                                const int*   __restrict__ recvs,
                                const float* __restrict__ conv,
                                float*       __restrict__ agg,
                                float*       __restrict__ csum,
                                int nEdges) {
    __shared__ int   s_send[CHUNK];
    __shared__ int   s_recv[CHUNK];
    __shared__ float s_conv[CHUNK];

    const int e0 = blockIdx.x * CHUNK;
    int n = nEdges - e0;
    if (n > CHUNK) n = CHUNK;
    const int t = threadIdx.x;
    if (t < n) {
        s_send[t] = senders[e0 + t];
        s_recv[t] = recvs[e0 + t];
        s_conv[t] = conv[e0 + t];
    }
    __syncthreads();
    if (n <= 0) return;

    const int col = t;
    float acc = 0.f;
    int   cur = s_recv[0];
    for (int i = 0; i < n; ++i) {
        const int r = s_recv[i];
        if (r != cur) {
            atomicAdd(&agg[(long long)cur * F + col], acc);
            acc = 0.f;
            cur = r;
        }
        acc = fmaf(s_conv[i], feats[(long long)s_send[i] * F + col], acc);
    }
    atomicAdd(&agg[(long long)cur * F + col], acc);

    if (t == 0) {
        float cs = 0.f;
        int   c2 = s_recv[0];
        for (int i = 0; i < n; ++i) {
            const int r = s_recv[i];
            if (r != c2) {
                atomicAdd(&csum[c2], cs);
                cs = 0.f;
                c2 = r;
            }
            cs += s_conv[i];
        }
        atomicAdd(&csum[c2], cs);
    }
}

// ---------------------------------------------------------------------------
// WMMA GEMM:  out[N_NODES,256] (=|*=) A[N_NODES,K] @ W[K,256] + csum[row]*bias
// Full f32 precision via V_WMMA_F32_16X16X4_F32 (16x16 tile, K-step 4).
// One wave computes a 16x64 tile (4 accumulators -> independent WMMAs).
//
// VGPR fragment layouts per CDNA5 ISA 7.12.2:
//   A 16x4 f32 : lanes 0-15 rows 0-15 hold K={0,1}; lanes 16-31 hold K={2,3}
//   B 4x16 f32 : VGPR0 lanes0-15 K=0 / lanes16-31 K=2; VGPR1 K=1 / K=3
//   C/D 16x16  : VGPR j: lanes0-15 row j, lanes16-31 row j+8; col = lane&15
// ---------------------------------------------------------------------------
template <int K, int LDA, bool MUL>
__global__ void wmma_gemm_kernel(const float* __restrict__ A,
                                 const float* __restrict__ W,     // [K,256]
                                 const float* __restrict__ bias,  // [256]
                                 const float* __restrict__ csum,  // [N_NODES]
                                 float*       __restrict__ out) { // [N_NODES,256]
    const int lane = threadIdx.x & 31;
    const int wid  = threadIdx.x >> 5;
    const int gw   = blockIdx.x * (blockDim.x >> 5) + wid;

    const int NGROUP = 4;               // 4 tiles of 16 cols = 64 cols / wave
    const int MT     = N_NODES / 16;    // 3125 row tiles (exact)
    if (gw >= MT * NGROUP) return;      // wave-uniform: EXEC stays all-1s

    const int m0 = (gw / NGROUP) * 16;
    const int n0 = (gw % NGROUP) * 64;
    const int hi = lane >> 4;           // half-wave selector
    const int l  = lane & 15;

    v8f acc[4];
#pragma unroll
    for (int tN = 0; tN < 4; ++tN) acc[tN] = 0.f;

    const float* Arow = A + (long long)(m0 + l) * LDA;
    for (int k0 = 0; k0 < K; k0 += 4) {
        const int kk = k0 + 2 * hi;
        v2f a;
        a.x = Arow[kk];
        a.y = Arow[kk + 1];
#pragma unroll
        for (int tN = 0; tN < 4; ++tN) {
            const int nc = n0 + tN * 16 + l;
            v2f b;
            b.x = W[(long long)kk * 256 + nc];
            b.y = W[(long long)(kk + 1) * 256 + nc];
            acc[tN] = __builtin_amdgcn_wmma_f32_16x16x4_f32(
                false, a, false, b, (short)0, acc[tN], false, false);
        }
    }

    // Epilogue: + csum[row]*bias[col]; optionally fuse Hadamard with out.
#pragma unroll
    for (int j = 0; j < 8; ++j) {
        const int   row = m0 + 8 * hi + j;
        const float cs  = csum[row];
#pragma unroll
        for (int tN = 0; tN < 4; ++tN) {
            const int  colI = n0 + tN * 16 + l;
            float v = acc[tN][j] + cs * bias[colI];
            const long long idx = (long long)row * 256 + colI;
            if (MUL) v *= out[idx];
            out[idx] = v;
        }
    }
}

// ---------------------------------------------------------------------------
extern "C" void kernel_launch(void* const* d_in, const int* in_sizes, int n_in,
                              void* d_out, int out_size, void* d_ws, size_t ws_size,
                              hipStream_t stream) {
    const float* node_features  = (const float*)d_in[0];
    const float* hedge_features = (const float*)d_in[1];
    const int*   node_senders   = (const int*)d_in[2];
    const int*   node_receivers = (const int*)d_in[3];
    const float* node_conv      = (const float*)d_in[4];
    const int*   h2n_senders    = (const int*)d_in[5];
    const int*   h2n_receivers  = (const int*)d_in[6];
    const float* h2n_conv       = (const float*)d_in[7];
    const float* W_msg          = (const float*)d_in[8];
    const float* b_msg          = (const float*)d_in[9];
    const float* W_scale        = (const float*)d_in[10];
    const float* b_scale        = (const float*)d_in[11];
    float*       out            = (float*)d_out;

    // Workspace layout (all float): agg1 | agg2 | csum1 | csum2
    float* agg1  = (float*)d_ws;
    float* agg2  = agg1 + (size_t)N_NODES * F_NODE;
    float* csum1 = agg2 + (size_t)N_NODES * F_HEDGE;
    float* csum2 = csum1 + N_NODES;

    const size_t zeroN = (size_t)N_NODES * F_NODE +
                         (size_t)N_NODES * F_HEDGE + 2 * (size_t)N_NODES;
    const int n4 = (int)(zeroN / 4);
    zero_kernel<<<(n4 + 255) / 256, 256, 0, stream>>>((float4*)d_ws, n4);

    // Segment-sum of scaled sender features (node->node, then hedge->node).
    edge_agg_kernel<F_NODE><<<(E1 + CHUNK - 1) / CHUNK, F_NODE, 0, stream>>>(
        node_features, node_senders, node_receivers, node_conv, agg1, csum1, E1);
    edge_agg_kernel<F_HEDGE><<<(E2 + CHUNK - 1) / CHUNK, F_HEDGE, 0, stream>>>(
        hedge_features, h2n_senders, h2n_receivers, h2n_conv, agg2, csum2, E2);

    // Projection GEMMs on aggregated features (13x fewer FLOPs than per-edge).
    const int waves  = (N_NODES / 16) * 4;   // 12500
    const int blocks = (waves + 7) / 8;      // 8 waves / 256-thread block
    wmma_gemm_kernel<256, 256, false><<<blocks, 256, 0, stream>>>(
        agg1, W_msg, b_msg, csum1, out);
    wmma_gemm_kernel<128, 128, true><<<blocks, 256, 0, stream>>>(
        agg2, W_scale, b_scale, csum2, out);

    (void)in_sizes; (void)n_in; (void)out_size; (void)ws_size;
}